// BranchRoute_60284160966844
// MI455X (gfx1250) — compile-verified
//
#include <hip/hip_runtime.h>

#define N_TOK 32768
#define DMODEL 2048

typedef float v2f __attribute__((ext_vector_type(2)));
typedef float v4f __attribute__((ext_vector_type(4)));
typedef float v8f __attribute__((ext_vector_type(8)));

// ---------------------------------------------------------------------------
// Kernel 1: gate GEMM via V_WMMA_F32_16X16X4_F32 (one wave = 16 token rows),
// emits per-tile mask words + packed counts, and streams combined = x*(m0+m1).
// W's 2 real columns are replicated across the 16 WMMA N-columns (col = lane&1)
// so every lane feeds valid data and EXEC stays all-ones (WMMA requirement).
// Output stream uses non-temporal stores: it is never re-read, while x must
// stay resident in the 192MB L2 for the later rank-gather.
// ---------------------------------------------------------------------------
__global__ __launch_bounds__(256) void gate_kernel(
    const float* __restrict__ x, const float* __restrict__ W,
    const float* __restrict__ bias, float* __restrict__ combined,
    unsigned* __restrict__ flags, unsigned* __restrict__ packedCnt)
{
    __shared__ float sW[DMODEL * 2];   // 16 KB: whole W staged once per block
    const int tid = threadIdx.x;
    for (int i = tid * 4; i < DMODEL * 2; i += 256 * 4)
        *(v4f*)&sW[i] = *(const v4f*)&W[i];
    __syncthreads();

    const int lane   = tid & 31;
    const int waveId = tid >> 5;
    const int tile   = blockIdx.x * 8 + waveId;   // 2048 tiles of 16 tokens
    const int rowBase = tile * 16;
    const int kSel   = lane >> 4;                 // 0: K=0,1  1: K=2,3 (A layout)
    const int colb   = lane & 1;                  // replicated W column

    const float* xrow = x + (size_t)(rowBase + (lane & 15)) * DMODEL + kSel * 2;

    v8f acc = {0.f, 0.f, 0.f, 0.f, 0.f, 0.f, 0.f, 0.f};
#pragma unroll 4
    for (int k = 0; k < DMODEL; k += 4) {
        v2f a = *(const v2f*)(xrow + k);          // A: 16x4 f32 fragment
        int ks = k + kSel * 2;
        v2f bf;                                   // B: 4x16 f32 fragment (cols = W col (lane&1))
        bf.x = sW[ks * 2 + colb];
        bf.y = sW[ks * 2 + 2 + colb];
        acc = __builtin_amdgcn_wmma_f32_16x16x4_f32(
            false, a, false, bf, (short)0, acc, false, false);
    }

    // D layout: VGPR j -> M = j + 8*(lane>=16), N = lane&15.
    // Path-p logits live in lanes p (M=0..7) and p+16 (M=8..15).
    float bv = bias[lane & 1];
    unsigned bits8 = 0;
#pragma unroll
    for (int j = 0; j < 8; ++j)
        bits8 |= (acc[j] + bv >= 0.f) ? (1u << j) : 0u;   // sigmoid(z)>=0.5 <=> z>=0

    unsigned w = (__shfl(bits8, 0, 32) & 0xFFu)
               | ((__shfl(bits8, 16, 32) & 0xFFu) << 8)
               | ((__shfl(bits8, 1, 32) & 0xFFu) << 16)
               | ((__shfl(bits8, 17, 32) & 0xFFu) << 24);

    if (lane == 0) {
        flags[tile] = w;
        packedCnt[tile] = __popc(w & 0xFFFFu) | (__popc(w >> 16) << 16);
    }

    // combined = x * (m0 + m1), coalesced float4 streaming over the 16 rows
    for (int m = 0; m < 16; ++m) {
        float s = (float)(((w >> m) & 1u) + ((w >> (16 + m)) & 1u));
        const v4f* src = (const v4f*)(x + (size_t)(rowBase + m) * DMODEL);
        v4f* dst = (v4f*)(combined + (size_t)(rowBase + m) * DMODEL);
        for (int c = lane; c < DMODEL / 4; c += 32) {
            v4f v = src[c] * s;
            __builtin_nontemporal_store(v, dst + c);
        }
    }
}

// ---------------------------------------------------------------------------
// Kernel 2: exclusive scan over 2048 packed tile counts (both paths in one u32;
// each 16-bit half never exceeds 32768, so no carry crossover). Single block.
// ---------------------------------------------------------------------------
__global__ __launch_bounds__(256) void scan_kernel(
    const unsigned* __restrict__ packedCnt, unsigned* __restrict__ packedOff,
    unsigned* __restrict__ totals)
{
    __shared__ unsigned sdata[256];
    const int tid = threadIdx.x;
    unsigned v[8], sum = 0;
#pragma unroll
    for (int j = 0; j < 8; ++j) { v[j] = packedCnt[tid * 8 + j]; sum += v[j]; }
    sdata[tid] = sum;
    __syncthreads();
    for (int d = 1; d < 256; d <<= 1) {           // Hillis-Steele inclusive
        unsigned t = (tid >= d) ? sdata[tid - d] : 0u;
        __syncthreads();
        sdata[tid] += t;
        __syncthreads();
    }
    unsigned excl = sdata[tid] - sum;
#pragma unroll
    for (int j = 0; j < 8; ++j) { packedOff[tid * 8 + j] = excl; excl += v[j]; }
    if (tid == 255) totals[0] = sdata[255];
}

// ---------------------------------------------------------------------------
// Kernel 3: rank -> token gather lists (token order == jnp.cumsum order).
// ---------------------------------------------------------------------------
__global__ __launch_bounds__(256) void index_kernel(
    const unsigned* __restrict__ flags, const unsigned* __restrict__ packedOff,
    int* __restrict__ idx0, int* __restrict__ idx1)
{
    int t = blockIdx.x * 256 + threadIdx.x;
    unsigned w   = flags[t >> 4];
    unsigned off = packedOff[t >> 4];
    int bit = t & 15;
    unsigned below = (1u << bit) - 1u;
    if ((w >> bit) & 1u)
        idx0[(off & 0xFFFFu) + __popc(w & 0xFFFFu & below)] = t;
    if ((w >> (16 + bit)) & 1u)
        idx1[(off >> 16) + __popc((w >> 16) & below)] = t;
}

// ---------------------------------------------------------------------------
// Kernel 4: build x0/x1 via gather; rows >= count get zeros (capacity padding).
// One block per (row, path); 256 threads x 2 float4 = 2048 floats, coalesced.
// Gather loads stay regular-temporal (x rows are read once per routed path and
// benefit from L2); output stores are non-temporal (write-once streams).
// ---------------------------------------------------------------------------
__global__ __launch_bounds__(256) void emit_kernel(
    const float* __restrict__ x, const int* __restrict__ idx0,
    const int* __restrict__ idx1, const unsigned* __restrict__ totals,
    float* __restrict__ out)
{
    const int r = blockIdx.x, p = blockIdx.y, tid = threadIdx.x;
    unsigned tot = totals[0];
    unsigned cnt = p ? (tot >> 16) : (tot & 0xFFFFu);
    v4f* dst = (v4f*)(out + ((size_t)p * N_TOK + r) * DMODEL);
    if ((unsigned)r < cnt) {
        int tok = p ? idx1[r] : idx0[r];
        const v4f* src = (const v4f*)(x + (size_t)tok * DMODEL);
        __builtin_nontemporal_store(src[tid], dst + tid);
        __builtin_nontemporal_store(src[tid + 256], dst + tid + 256);
    } else {
        v4f z = {0.f, 0.f, 0.f, 0.f};
        __builtin_nontemporal_store(z, dst + tid);
        __builtin_nontemporal_store(z, dst + tid + 256);
    }
}

extern "C" void kernel_launch(void* const* d_in, const int* in_sizes, int n_in,
                              void* d_out, int out_size, void* d_ws, size_t ws_size,
                              hipStream_t stream) {
    (void)in_sizes; (void)n_in; (void)out_size; (void)ws_size;
    const float* x = (const float*)d_in[0];
    const float* W = (const float*)d_in[1];
    const float* b = (const float*)d_in[2];
    float* out      = (float*)d_out;                             // x0 | x1 | combined
    float* combined = out + (size_t)2 * N_TOK * DMODEL;

    char* ws = (char*)d_ws;
    unsigned* flags     = (unsigned*)(ws);            // 2048 * 4
    unsigned* packedCnt = (unsigned*)(ws + 8192);     // 2048 * 4
    unsigned* packedOff = (unsigned*)(ws + 16384);    // 2048 * 4
    unsigned* totals    = (unsigned*)(ws + 24576);    // 256 B slot
    int* idx0           = (int*)(ws + 24832);         // N * 4
    int* idx1           = (int*)(ws + 24832 + (size_t)N_TOK * 4);

    gate_kernel <<<N_TOK / 128, 256, 0, stream>>>(x, W, b, combined, flags, packedCnt);
    scan_kernel <<<1, 256, 0, stream>>>(packedCnt, packedOff, totals);
    index_kernel<<<N_TOK / 256, 256, 0, stream>>>(flags, packedOff, idx0, idx1);
    emit_kernel <<<dim3(N_TOK, 2), 256, 0, stream>>>(x, idx0, idx1, totals, out);
}